// FastRCNNOutputLayers_44650480009336
// MI455X (gfx1250) — compile-verified
//
#include <hip/hip_runtime.h>
#include <stdint.h>

// ---------------- problem constants (match reference) ----------------
#define RR   400
#define KK   20
#define NCAND (RR * KK)          // 8000 candidates
#define NPAD  8192               // padded to power of two for bitonic sort
#define NTH   1024               // one workgroup, 32 wave32 waves
#define SCORE_THRESH 0.05f
#define NMS_THRESH   0.5f
#define TOPK  100
#define OUTC  (4 + 1 + KK)       // 25 output columns

// ---------------- LDS layout (total ~206 KB < 320 KB WGP LDS) ---------
// The raw AoS boxes (128000 B) are staged by the TDM into the region that
// later becomes the SoA X1/Y1/X2/Y2 arrays; raw scores (33600 B) are staged
// into the region that later becomes SKEY/SPAY. Barriers separate the
// staging reads from the SoA writes.
#define OFF_X1   0                        // 8192 f32
#define OFF_Y1   (OFF_X1 + 32768)
#define OFF_X2   (OFF_Y1 + 32768)
#define OFF_Y2   (OFF_X2 + 32768)        // ends 131072
#define OFF_SKEY 131072                   // 8192 f32 sorted scores
#define OFF_SPAY (OFF_SKEY + 32768)       // 8192 i32 payload (idx | valid<<20)
#define OFF_SUPP (OFF_SPAY + 32768)       // 8192 u8 suppression flags
#define OFF_RED  (OFF_SUPP + 8192)        // 1024 f32 reduction scratch
#define OFF_SEL  (OFF_RED + 4096)         // 128 i32 selected positions
#define OFF_MISC (OFF_SEL + 512)          // 16 i32 misc (kcount, max_coord+1)
#define OFF_BLK  (OFF_MISC + 64)          // 5*32 f32 block boxes + 32 i32 kept
#define OFF_RFIN (OFF_BLK + 768)          // 400 u8 per-row finite flags
#define LDS_TOTAL (OFF_RFIN + 512)        // 210 752 B

// ---------------- TDM descriptor types (6-arg builtin toolchain) -------
typedef unsigned int u32x4 __attribute__((ext_vector_type(4)));
typedef int          i32x8 __attribute__((ext_vector_type(8)));
typedef int          i32x4 __attribute__((ext_vector_type(4)));

// 1-D DMA of `nelem` 32-bit elements global -> LDS via the Tensor Data Mover.
// D# built per CDNA5 ISA §8.3 (group0) / §8.4 (group1); groups 2/3 unused
// (tile is a single row), so the trailing descriptor operands are zero.
__device__ __forceinline__ void tdm_load_1d(const void* gptr, void* lptr, int nelem) {
    uint64_t ga = (uint64_t)(uintptr_t)gptr;
    uint32_t la = (uint32_t)(uintptr_t)lptr;   // low 32 bits of LDS generic addr = byte offset
    u32x4 g0;
    g0[0] = 1u;                                          // count=1, user mode, no gather
    g0[1] = la;                                          // lds_addr
    g0[2] = (uint32_t)(ga & 0xFFFFFFFFu);                // global_addr[31:0]
    g0[3] = (uint32_t)((ga >> 32) & 0x1FFFFFFu)          // global_addr[56:32]
          | (2u << 30);                                  // type = 2 ("image")
    i32x8 g1;
    g1[0] = (int)(2u << 16);                             // wg_mask=0, data_size=2 (4B)
    g1[1] = (int)((nelem & 0xFFFF) << 16);               // tensor_dim0[15:0]
    g1[2] = (int)(((nelem >> 16) & 0xFFFF) | (1 << 16)); // tensor_dim0[31:16] | tensor_dim1=1
    g1[3] = (int)((nelem & 0xFFFF) << 16);               // tile_dim0 = nelem (<=65535)
    g1[4] = 1;                                           // tile_dim1=1, tile_dim2=0
    g1[5] = nelem;                                       // tensor_dim0_stride lo32
    g1[6] = 0;
    g1[7] = 0;
    i32x4 z4 = {0, 0, 0, 0};
    i32x8 z8 = {0, 0, 0, 0, 0, 0, 0, 0};
    __builtin_amdgcn_tensor_load_to_lds(g0, g1, z4, z4, z8, 0);
}

__device__ __forceinline__ float clip_keep_nan(float v, float lo, float hi) {
    // matches jnp.clip: NaN propagates, +/-inf clamps to bounds
    v = (v < lo) ? lo : v;
    v = (v > hi) ? hi : v;
    return v;
}

__global__ __launch_bounds__(NTH)
void fastrcnn_post_kernel(const float* __restrict__ boxes,
                          const float* __restrict__ scores,
                          const float* __restrict__ pred,
                          const int* __restrict__ imh_p,
                          const int* __restrict__ imw_p,
                          float* __restrict__ out) {
    extern __shared__ char smem[];
    float*         X1   = (float*)(smem + OFF_X1);
    float*         Y1   = (float*)(smem + OFF_Y1);
    float*         X2   = (float*)(smem + OFF_X2);
    float*         Y2   = (float*)(smem + OFF_Y2);
    float*         SKEY = (float*)(smem + OFF_SKEY);
    int*           SPAY = (int*)(smem + OFF_SPAY);
    unsigned char* SUPP = (unsigned char*)(smem + OFF_SUPP);
    float*         RED  = (float*)(smem + OFF_RED);
    int*           SEL  = (int*)(smem + OFF_SEL);
    int*           MISC = (int*)(smem + OFF_MISC);
    float*         MISCF= (float*)(smem + OFF_MISC);
    float*         BX1  = (float*)(smem + OFF_BLK);
    float*         BY1  = BX1 + 32;
    float*         BX2  = BY1 + 32;
    float*         BY2  = BX2 + 32;
    float*         BA   = BY2 + 32;
    int*           KBUF = (int*)(BA + 32);
    unsigned char* RFIN = (unsigned char*)(smem + OFF_RFIN);

    const float* stageBoxes  = (const float*)(smem + OFF_X1);   // raw AoS boxes
    const float* stageScores = (const float*)(smem + OFF_SKEY); // raw R x (K+1)

    const int tid  = (int)threadIdx.x;
    const int lane = tid & 31;

    // ---- S1: TDM async loads into LDS (issued once by wave 0) ----
    if (tid == 0) {
        tdm_load_1d(boxes,  smem + OFF_X1,   NCAND * 4);       // 32000 f32
        tdm_load_1d(scores, smem + OFF_SKEY, RR * (KK + 1));   // 8400 f32
        __builtin_amdgcn_s_wait_tensorcnt(0);
    }
    for (int e = tid; e < NPAD; e += NTH) SUPP[e] = 0;
    __syncthreads();

    // ---- S2: per-row "all scores finite" flags ----
    if (tid < RR) {
        bool f = true;
        for (int k = 0; k < KK; ++k)
            f = f && __builtin_isfinite(stageScores[tid * (KK + 1) + k]);
        RFIN[tid] = f ? 1 : 0;
    }
    __syncthreads();

    const float imW = (float)imw_p[0];
    const float imH = (float)imh_p[0];

    // ---- S3: clip + validity into registers (staging still live) ----
    float rx1[NPAD / NTH], ry1[NPAD / NTH], rx2[NPAD / NTH], ry2[NPAD / NTH], rsc[NPAD / NTH];
    int   rvalid[NPAD / NTH];
    float lmax = -3.0e38f;
#pragma unroll
    for (int it = 0; it < NPAD / NTH; ++it) {
        int c = tid + it * NTH;
        if (c < NCAND) {
            float b0 = stageBoxes[c * 4 + 0], b1 = stageBoxes[c * 4 + 1];
            float b2 = stageBoxes[c * 4 + 2], b3 = stageBoxes[c * 4 + 3];
            int r = c / KK, k = c - r * KK;
            float sc = stageScores[r * (KK + 1) + k];
            float cx1 = clip_keep_nan(b0, 0.f, imW);
            float cy1 = clip_keep_nan(b1, 0.f, imH);
            float cx2 = clip_keep_nan(b2, 0.f, imW);
            float cy2 = clip_keep_nan(b3, 0.f, imH);
            bool fin = __builtin_isfinite(cx1) && __builtin_isfinite(cy1) &&
                       __builtin_isfinite(cx2) && __builtin_isfinite(cy2) &&
                       (RFIN[r] != 0);
            rvalid[it] = (sc > SCORE_THRESH && fin) ? 1 : 0;
            rx1[it] = cx1; ry1[it] = cy1; rx2[it] = cx2; ry2[it] = cy2; rsc[it] = sc;
            lmax = fmaxf(lmax, fmaxf(fmaxf(cx1, cy1), fmaxf(cx2, cy2)));
        } else {
            rvalid[it] = -1;  // padding slot
        }
    }
    __syncthreads();   // all staging reads complete

    // ---- S4: write SoA boxes + sort keys/payloads (overwrites staging) ----
#pragma unroll
    for (int it = 0; it < NPAD / NTH; ++it) {
        int c = tid + it * NTH;
        if (rvalid[it] >= 0) {
            X1[c] = rx1[it]; Y1[c] = ry1[it]; X2[c] = rx2[it]; Y2[c] = ry2[it];
            SKEY[c] = rsc[it];
            SPAY[c] = c | (rvalid[it] ? (1 << 20) : 0);
        } else {
            X1[c] = Y1[c] = X2[c] = Y2[c] = 0.f;
            SKEY[c] = -__builtin_inff();
            SPAY[c] = 0xFFFF;               // pad: huge idx, valid bit clear
        }
    }
    RED[tid] = lmax;
    __syncthreads();

    // ---- S5: max coordinate reduction -> class offset scale ----
    if (tid == 0) {
        float m = RED[0];
        for (int i = 1; i < NTH; ++i) m = fmaxf(m, RED[i]);
        MISCF[2] = m + 1.0f;                // (max_coord + 1)
    }
    __syncthreads();
    const float mc1 = MISCF[2];

    // ---- S6: bitonic sort of 8192 (score desc, index asc) in LDS ----
    for (int k2 = 2; k2 <= NPAD; k2 <<= 1) {
        for (int j = k2 >> 1; j > 0; j >>= 1) {
            for (int t = tid; t < NPAD; t += NTH) {
                int ixj = t ^ j;
                if (ixj > t) {
                    float ka = SKEY[t], kb = SKEY[ixj];
                    int   pa = SPAY[t], pb = SPAY[ixj];
                    bool pre = (ka > kb) || (ka == kb && pa < pb); // a precedes b (desc)
                    bool dirDesc = ((t & k2) == 0);
                    if (dirDesc ? !pre : pre) {
                        SKEY[t] = kb; SKEY[ixj] = ka;
                        SPAY[t] = pb; SPAY[ixj] = pa;
                    }
                }
            }
            __syncthreads();
        }
    }

    // ---- S7: blocked greedy NMS (block = 32 = one wave32) ----
    for (int b = 0; b < NCAND / 32; ++b) {
        const int base = b * 32;
        if (tid < 32) {   // wave 0: sequential resolve within block via shuffles
            int e = base + lane;
            int pay = SPAY[e];
            int valid = (pay >> 20) & 1;
            int idx = pay & 0xFFFF; if (idx >= NCAND) idx = 0;
            int cls = idx % KK;
            float off = (float)cls * mc1;
            float ax1 = X1[idx] + off, ay1 = Y1[idx] + off;
            float ax2 = X2[idx] + off, ay2 = Y2[idx] + off;
            float aarea = (ax2 - ax1) * (ay2 - ay1);
            int sup = (int)SUPP[e];
            int keepL = 0;
            for (int i = 0; i < 32; ++i) {
                int flag = (valid && !sup) ? 1 : 0;
                int   ki = __shfl(flag,  i, 32);
                float bx1 = __shfl(ax1,  i, 32);
                float by1 = __shfl(ay1,  i, 32);
                float bx2 = __shfl(ax2,  i, 32);
                float by2 = __shfl(ay2,  i, 32);
                float bar = __shfl(aarea, i, 32);
                if (ki) {
                    if (lane == i) keepL = 1;
                    else if (lane > i) {
                        float iw = fminf(ax2, bx2) - fmaxf(ax1, bx1); iw = iw > 0.f ? iw : 0.f;
                        float ih = fminf(ay2, by2) - fmaxf(ay1, by1); ih = ih > 0.f ? ih : 0.f;
                        float inter = iw * ih;
                        float iou = inter / (aarea + bar - inter + 1e-12f);
                        if (iou > NMS_THRESH) sup = 1;
                    }
                }
            }
            SUPP[e] = (unsigned char)sup;
            BX1[lane] = ax1; BY1[lane] = ay1; BX2[lane] = ax2; BY2[lane] = ay2;
            BA[lane] = aarea; KBUF[lane] = keepL;
        }
        __syncthreads();
        // all 32 waves: propagate suppression from this block's kept boxes
        for (int e2 = base + 32 + tid; e2 < NCAND; e2 += NTH) {
            if (SUPP[e2]) continue;
            int pay2 = SPAY[e2];
            int idx2 = pay2 & 0xFFFF; if (idx2 >= NCAND) idx2 = 0;
            int cls2 = idx2 % KK;
            float off2 = (float)cls2 * mc1;
            float cxx1 = X1[idx2] + off2, cyy1 = Y1[idx2] + off2;
            float cxx2 = X2[idx2] + off2, cyy2 = Y2[idx2] + off2;
            float carea = (cxx2 - cxx1) * (cyy2 - cyy1);
            int s = 0;
            for (int i = 0; i < 32 && !s; ++i) {
                if (!KBUF[i]) continue;
                float iw = fminf(cxx2, BX2[i]) - fmaxf(cxx1, BX1[i]); iw = iw > 0.f ? iw : 0.f;
                float ih = fminf(cyy2, BY2[i]) - fmaxf(cyy1, BY1[i]); ih = ih > 0.f ? ih : 0.f;
                float inter = iw * ih;
                float iou = inter / (carea + BA[i] - inter + 1e-12f);
                if (iou > NMS_THRESH) s = 1;
            }
            if (s) SUPP[e2] = 1;
        }
        __syncthreads();
    }

    // ---- S8: first 100 kept in sorted order == top_k of keep?score:-inf ----
    if (tid == 0) {
        int cnt = 0;
        for (int e = 0; e < NCAND && cnt < TOPK; ++e) {
            int pay = SPAY[e];
            if (((pay >> 20) & 1) && !SUPP[e]) SEL[cnt++] = e;
        }
        MISC[0] = cnt;
    }
    __syncthreads();
    const int kcount = MISC[0];

    // ---- S9: gather outputs ----
    if (tid < TOPK) {
        int m = tid;
        float* orow   = out + m * OUTC;
        int*   ocls   = (int*)out + TOPK * OUTC;   // after 2500 floats
        int*   orowid = ocls + TOPK;
        if (m < kcount) {
            int e = SEL[m];
            int pay = SPAY[e];
            int idx = pay & 0xFFFF;
            int r = idx / KK, cls = idx - r * KK;
            orow[0] = X1[idx]; orow[1] = Y1[idx]; orow[2] = X2[idx]; orow[3] = Y2[idx];
            orow[4] = SKEY[e];
            for (int q = 0; q < KK; ++q) orow[5 + q] = pred[r * (KK + 1) + q];
            ocls[m] = cls; orowid[m] = r;
        } else {
            for (int q = 0; q < OUTC; ++q) orow[q] = 0.f;
            ocls[m] = -1; orowid[m] = -1;
        }
    }
}

extern "C" void kernel_launch(void* const* d_in, const int* in_sizes, int n_in,
                              void* d_out, int out_size, void* d_ws, size_t ws_size,
                              hipStream_t stream) {
    (void)in_sizes; (void)n_in; (void)out_size; (void)d_ws; (void)ws_size;
    const float* boxes  = (const float*)d_in[0];
    const float* scores = (const float*)d_in[1];
    const float* pred   = (const float*)d_in[2];
    const int*   imh    = (const int*)d_in[3];
    const int*   imw    = (const int*)d_in[4];
    float* out = (float*)d_out;

    // allow >64KB dynamic LDS (gfx1250 WGP has 320KB); deterministic, host-side only
    (void)hipFuncSetAttribute((const void*)fastrcnn_post_kernel,
                              hipFuncAttributeMaxDynamicSharedMemorySize, LDS_TOTAL);

    fastrcnn_post_kernel<<<dim3(1), dim3(NTH), LDS_TOTAL, stream>>>(
        boxes, scores, pred, imh, imw, out);
}